// Encoding_Layer_84825604096013
// MI455X (gfx1250) — compile-verified
//
#include <hip/hip_runtime.h>

typedef __attribute__((ext_vector_type(2))) float v2f;
typedef __attribute__((ext_vector_type(8))) float v8f;

namespace {
constexpr int Bsz = 32;
constexpr int Tsz = 512;
constexpr int Dsz = 512;
constexpr int Hsz = 512;
constexpr int NGsz = 2 * Hsz;          // 1024 gate columns
constexpr int NWG = 16;                // persistent blocks
constexpr int THREADS = 256;           // 8 waves/block -> 128 waves total

constexpr size_t OFF_H  = 256;
constexpr size_t OFF_RH = OFF_H  + (size_t)Bsz * Hsz * sizeof(float);
constexpr size_t OFF_U  = OFF_RH + (size_t)Bsz * Hsz * sizeof(float);
constexpr size_t WS_BYTES = OFF_U + (size_t)Bsz * Hsz * sizeof(float);
}

__device__ __forceinline__ v8f wmma4(v2f a, v2f b, v8f c) {
  // V_WMMA_F32_16X16X4_F32 : D = A(16x4) * B(4x16) + C, full fp32
  return __builtin_amdgcn_wmma_f32_16x16x4_f32(false, a, false, b, (short)0, c,
                                               false, false);
}

__device__ __forceinline__ float sigm(float x) { return 1.0f / (1.0f + expf(-x)); }

__device__ __forceinline__ void grid_barrier(unsigned* cnt, unsigned target) {
  __threadfence();
  __syncthreads();
  if (threadIdx.x == 0) {
    __hip_atomic_fetch_add(cnt, 1u, __ATOMIC_RELEASE, __HIP_MEMORY_SCOPE_AGENT);
    while (__hip_atomic_load(cnt, __ATOMIC_ACQUIRE, __HIP_MEMORY_SCOPE_AGENT) < target) {
      __builtin_amdgcn_s_sleep(1);
    }
  }
  __syncthreads();
  __threadfence();
}

// K-panel MAC: A row-major [rows x ld not used: row ptr given], W row-major with
// leading dim ldw, columns fixed per lane. Two independent accumulator chains
// (even/odd k-step) so consecutive v_wmma ops have no RAW on the accumulator.
__device__ __forceinline__ void kpanel(const float* __restrict__ arow,
                                       const float* __restrict__ w,
                                       int ldw, int kHalf,
                                       v8f& acc0, v8f& acc1) {
  #pragma unroll 8
  for (int k = 0; k < 512; k += 8) {
    v2f a0 = *(const v2f*)(arow + k + kHalf);
    v2f a1 = *(const v2f*)(arow + k + 4 + kHalf);
    v2f b0, b1;
    b0.x = w[(k + kHalf) * ldw];
    b0.y = w[(k + kHalf + 1) * ldw];
    b1.x = w[(k + 4 + kHalf) * ldw];
    b1.y = w[(k + 4 + kHalf + 1) * ldw];
    acc0 = wmma4(a0, b0, acc0);
    acc1 = wmma4(a1, b1, acc1);
  }
}

__global__ __launch_bounds__(THREADS) void bigru_last_kernel(
    const float* __restrict__ x,
    const float* __restrict__ fgk, const float* __restrict__ fgb,
    const float* __restrict__ fck, const float* __restrict__ fcb,
    const float* __restrict__ bgk, const float* __restrict__ bgb,
    const float* __restrict__ bck, const float* __restrict__ bcb,
    const int*   __restrict__ len,
    float* __restrict__ out,
    unsigned char* __restrict__ ws)
{
  unsigned* cnt = (unsigned*)ws;
  float* h  = (float*)(ws + OFF_H);    // [32][512] hidden state
  float* rh = (float*)(ws + OFF_RH);   // [32][512] r * h
  float* uu = (float*)(ws + OFF_U);    // [32][512] update gate

  const int lane  = threadIdx.x & 31;
  const int wave  = threadIdx.x >> 5;
  const int gw    = blockIdx.x * (THREADS / 32) + wave;   // 0..127

  const int mIn   = lane & 15;               // A row / B,C column within tile
  const int kHalf = (lane >> 4) << 1;        // K sub-offset for A/B: 0 or 2
  const int mBase = (lane >> 4) << 3;        // C row base: 0 or 8

  // gate tile: 128 tiles = 2 row-tiles x 64 col-tiles, one per wave
  const int gmt = gw >> 6;
  const int gn0 = (gw & 63) << 4;
  const int gRow = (gmt << 4) + mIn;         // batch row for A loads
  const int gnB  = gn0 + mIn;                // gate column for B loads / C cols

  // candidate tile: 64 tiles = 2 x 32, waves 0..63
  const int cmt = (gw >> 5) & 1;
  const int cn0 = (gw & 31) << 4;
  const int cRow = (cmt << 4) + mIn;
  const int cnB  = cn0 + mIn;

  unsigned phase = 0;

  for (int t = 0; t < Tsz; ++t) {
    // ---------------- gate GEMM: concat(x_t, h) @ fgk ----------------
    v8f acc0 = {};
    v8f acc1 = {};
    {
      const float* xrow = x + ((size_t)gRow * Tsz + t) * Dsz;
      const float* hrow = h + gRow * Hsz;
      kpanel(xrow, fgk + gnB,              NGsz, kHalf, acc0, acc1);
      kpanel(hrow, fgk + Dsz * NGsz + gnB, NGsz, kHalf, acc0, acc1);
    }
    {
      const float bias = fgb[gnB];
      if (gn0 < Hsz) {            // r gate -> stage r*h (wave-uniform branch)
        #pragma unroll
        for (int i = 0; i < 8; ++i) {
          int m = (gmt << 4) + mBase + i;
          float r = sigm(acc0[i] + acc1[i] + bias);
          rh[m * Hsz + gnB] = r * h[m * Hsz + gnB];
        }
      } else {                    // u gate
        int j = gnB - Hsz;
        #pragma unroll
        for (int i = 0; i < 8; ++i) {
          int m = (gmt << 4) + mBase + i;
          uu[m * Hsz + j] = sigm(acc0[i] + acc1[i] + bias);
        }
      }
    }
    ++phase; grid_barrier(cnt, phase * NWG);

    // ---------------- candidate GEMM: concat(x_t, r*h) @ fck ----------------
    if (gw < 64) {
      v8f ac0 = {};
      v8f ac1 = {};
      const float* xrow = x + ((size_t)cRow * Tsz + t) * Dsz;
      const float* rrow = rh + cRow * Hsz;
      kpanel(xrow, fck + cnB,             Hsz, kHalf, ac0, ac1);
      kpanel(rrow, fck + Dsz * Hsz + cnB, Hsz, kHalf, ac0, ac1);
      const float bias = fcb[cnB];
      #pragma unroll
      for (int i = 0; i < 8; ++i) {
        int m = (cmt << 4) + mBase + i;
        float c  = tanhf(ac0[i] + ac1[i] + bias);
        float uv = uu[m * Hsz + cnB];
        float ho = h[m * Hsz + cnB];
        h[m * Hsz + cnB] = uv * ho + (1.0f - uv) * c;  // unmasked update (exact for len==T)
      }
    }
    ++phase; grid_barrier(cnt, phase * NWG);
  }

  // ---------------- epilogue ----------------
  if (gw < 64) {
    // backward direction needs exactly one step from h=0 with input x[:,T-1,:]:
    // r is irrelevant (h=0), candidate's r*h rows are zero -> two K=512 GEMMs.
    v8f aU0 = {}, aU1 = {};
    v8f aC0 = {}, aC1 = {};
    const float* xrow = x + ((size_t)cRow * Tsz + (Tsz - 1)) * Dsz;
    kpanel(xrow, bgk + Hsz + cnB, NGsz, kHalf, aU0, aU1);
    kpanel(xrow, bck + cnB,       Hsz,  kHalf, aC0, aC1);
    const float biasU = bgb[Hsz + cnB];
    const float biasC = bcb[cnB];
    #pragma unroll
    for (int i = 0; i < 8; ++i) {
      int m = (cmt << 4) + mBase + i;
      float uv = sigm(aU0[i] + aU1[i] + biasU);
      float c  = tanhf(aC0[i] + aC1[i] + biasC);
      float hb = (1.0f - uv) * c;
      out[m * (2 * Hsz) + Hsz + cnB] = (len[m] == Tsz) ? hb : 0.0f;
    }
  } else {
    // forward half of the output: masked final hidden state
    const int f = gw - 64;                     // 0..63, 256 elements each
    #pragma unroll
    for (int j = 0; j < 8; ++j) {
      int idx = f * 256 + j * 32 + lane;       // coalesced
      int b = idx >> 9;
      int n = idx & (Hsz - 1);
      out[b * (2 * Hsz) + n] = (len[b] == Tsz) ? h[b * Hsz + n] : 0.0f;
    }
  }
}

extern "C" void kernel_launch(void* const* d_in, const int* in_sizes, int n_in,
                              void* d_out, int out_size, void* d_ws, size_t ws_size,
                              hipStream_t stream) {
  (void)in_sizes; (void)n_in; (void)out_size;
  const float* x   = (const float*)d_in[0];
  const float* fgk = (const float*)d_in[1];
  const float* fgb = (const float*)d_in[2];
  const float* fck = (const float*)d_in[3];
  const float* fcb = (const float*)d_in[4];
  const float* bgk = (const float*)d_in[5];
  const float* bgb = (const float*)d_in[6];
  const float* bck = (const float*)d_in[7];
  const float* bcb = (const float*)d_in[8];
  const int*   len = (const int*)d_in[9];

  size_t clear = WS_BYTES <= ws_size ? WS_BYTES : ws_size;
  hipMemsetAsync(d_ws, 0, clear, stream);   // zero barrier counter + h (h0 = 0)

  bigru_last_kernel<<<NWG, THREADS, 0, stream>>>(
      x, fgk, fgb, fck, fcb, bgk, bgb, bck, bcb, len,
      (float*)d_out, (unsigned char*)d_ws);
}